// CRFLayer_17523466568226
// MI455X (gfx1250) — compile-verified
//
#include <hip/hip_runtime.h>

#define Bc 128
#define Tc 1024
#define Nc 256

typedef unsigned int uint32;
typedef uint32 v4u __attribute__((ext_vector_type(4)));
typedef int    v4i __attribute__((ext_vector_type(4)));
typedef int    v8i __attribute__((ext_vector_type(8)));

// ---------------------------------------------------------------------------
// Tensor Data Mover: 2D tile (global -> LDS), descriptor per CDNA5 ISA.
//   elem_code: 0=1B, 1=2B, 2=4B, 3=8B ; dims/strides/tiles in data_size units
// ---------------------------------------------------------------------------
__device__ __forceinline__ void tdm_load_2d_to_lds(uint32 lds_byte_off,
                                                   const void* gptr,
                                                   uint32 elem_code,
                                                   uint32 tensor_d0, uint32 tensor_d1,
                                                   uint32 d0_stride,
                                                   uint32 tile_d0, uint32 tile_d1)
{
    unsigned long long ga = (unsigned long long)(size_t)gptr;
    v4u g0;
    g0[0] = 1u;                                      // count=1, user mode
    g0[1] = lds_byte_off;                            // lds_addr [63:32]
    g0[2] = (uint32)(ga & 0xFFFFFFFFull);            // global_addr [95:64]
    g0[3] = (uint32)((ga >> 32) & 0x1FFFFFFull)      // global_addr [120:96]
          | (2u << 30);                              // type=2 ("image") [127:126]
    v8i g1;
    g1[0] = (int)(elem_code << 16);                  // workgroup_mask=0, data_size
    g1[1] = (int)((tensor_d0 & 0xFFFFu) << 16);      // atomic_barrier_addr=0, dim0 lo
    g1[2] = (int)(((tensor_d0 >> 16) & 0xFFFFu) | ((tensor_d1 & 0xFFFFu) << 16));
    g1[3] = (int)(((tensor_d1 >> 16) & 0xFFFFu) | ((tile_d0 & 0xFFFFu) << 16));
    g1[4] = (int)(tile_d1 & 0xFFFFu);                // tile_dim1 ; tile_dim2=0
    g1[5] = (int)d0_stride;                          // tensor_dim0_stride [31:0]
    g1[6] = 0;
    g1[7] = 0;
    v4i gz = {0, 0, 0, 0};                           // groups 2/3 unused (2D)
#if defined(__clang_major__) && (__clang_major__ >= 23)
    v8i gz8 = {0, 0, 0, 0, 0, 0, 0, 0};
    __builtin_amdgcn_tensor_load_to_lds(g0, g1, gz, gz, gz8, 0);
#else
    __builtin_amdgcn_tensor_load_to_lds(g0, g1, gz, gz, 0);
#endif
}

// ---------------------------------------------------------------------------
// Kernel 1: forward Viterbi. One block per batch, 256 threads (8 wave32).
// Thread (jj = tid&63, q = tid>>6) owns columns 4jj..4jj+3, i in [64q,64q+64).
// LDS: [0,256KB) transitions (TDM), +1KB alpha, +4KB partial max, +4KB argmax.
// ---------------------------------------------------------------------------
__global__ __launch_bounds__(256) void viterbi_forward(
    const float* __restrict__ pot, const float* __restrict__ trans,
    const int* __restrict__ lengths, unsigned char* __restrict__ bp,
    int* __restrict__ lastTag)
{
    extern __shared__ char smem[];
    float* str    = (float*)smem;                           // 256 KB (row-major)
    float* salpha = (float*)(smem + 262144);                // 1 KB
    float* sredm  = (float*)(smem + 262144 + 1024);         // 4 KB partial max
    int*   sredi  = (int*)  (smem + 262144 + 1024 + 4096);  // 4 KB partial argmax

    const int b   = blockIdx.x;
    const int tid = threadIdx.x;
    const int jj  = tid & 63;                               // column group
    const int q   = tid >> 6;                               // i-quarter

    if (tid == 0)
        tdm_load_2d_to_lds((uint32)(size_t)str, trans, /*4B*/2, Nc, Nc, Nc, Nc, Nc);

    const size_t base = (size_t)b * Tc * Nc;
    const int len = lengths[b];

    salpha[tid] = pot[base + tid];                          // alpha0 = pot[:,0,:]

    __builtin_amdgcn_s_wait_tensorcnt(0);                   // wave0 waits TDM
    __syncthreads();

    const int i0 = q << 6;
    const float* trc = str + (jj << 2);                     // column-group base
    const int pbase = (q << 8) + (jj << 2);                 // partial slot

    for (int t = 1; t < len; ++t) {                         // len uniform per block
        float potc = pot[base + (size_t)t * Nc + tid];
        if (t + 1 < len)
            __builtin_prefetch(&pot[base + (size_t)(t + 1) * Nc + tid], 0, 0);

        float m0 = -3.402823466e38f, m1 = m0, m2 = m0, m3 = m0;
        int a0 = 0, a1 = 0, a2 = 0, a3 = 0;
        for (int i = i0; i < i0 + 64; i += 4) {
            float4 av = *(const float4*)(salpha + i);        // wave-broadcast b128
            float4 t0 = *(const float4*)(trc + (i + 0) * Nc); // contiguous b128
            float4 t1 = *(const float4*)(trc + (i + 1) * Nc);
            float4 t2 = *(const float4*)(trc + (i + 2) * Nc);
            float4 t3 = *(const float4*)(trc + (i + 3) * Nc);
            float s;
            s = av.x + t0.x; if (s > m0) { m0 = s; a0 = i;     }
            s = av.x + t0.y; if (s > m1) { m1 = s; a1 = i;     }
            s = av.x + t0.z; if (s > m2) { m2 = s; a2 = i;     }
            s = av.x + t0.w; if (s > m3) { m3 = s; a3 = i;     }
            s = av.y + t1.x; if (s > m0) { m0 = s; a0 = i + 1; }
            s = av.y + t1.y; if (s > m1) { m1 = s; a1 = i + 1; }
            s = av.y + t1.z; if (s > m2) { m2 = s; a2 = i + 1; }
            s = av.y + t1.w; if (s > m3) { m3 = s; a3 = i + 1; }
            s = av.z + t2.x; if (s > m0) { m0 = s; a0 = i + 2; }
            s = av.z + t2.y; if (s > m1) { m1 = s; a1 = i + 2; }
            s = av.z + t2.z; if (s > m2) { m2 = s; a2 = i + 2; }
            s = av.z + t2.w; if (s > m3) { m3 = s; a3 = i + 2; }
            s = av.w + t3.x; if (s > m0) { m0 = s; a0 = i + 3; }
            s = av.w + t3.y; if (s > m1) { m1 = s; a1 = i + 3; }
            s = av.w + t3.z; if (s > m2) { m2 = s; a2 = i + 3; }
            s = av.w + t3.w; if (s > m3) { m3 = s; a3 = i + 3; }
        }
        *(float4*)(sredm + pbase) = make_float4(m0, m1, m2, m3);
        *(int4*)  (sredi + pbase) = make_int4(a0, a1, a2, a3);
        __syncthreads();                                    // partials visible

        // combine: thread `tid` owns column c = tid; ascending q + strict '>'
        // preserves first-index argmax tie-breaking (matches jnp.argmax)
        float m = sredm[tid];
        int  am = sredi[tid];
        float v; int ai;
        v = sredm[256 + tid]; ai = sredi[256 + tid]; if (v > m) { m = v; am = ai; }
        v = sredm[512 + tid]; ai = sredi[512 + tid]; if (v > m) { m = v; am = ai; }
        v = sredm[768 + tid]; ai = sredi[768 + tid]; if (v > m) { m = v; am = ai; }

        bp[base + (size_t)t * Nc + tid] = (unsigned char)am;
        salpha[tid] = m + potc;
        __syncthreads();                                    // alpha/sred reusable
    }

    if (tid == 0) {                                         // last_tag = first-argmax
        float m = -3.402823466e38f;
        int am = 0;
        for (int i = 0; i < Nc; ++i) {
            float v2 = salpha[i];
            if (v2 > m) { m = v2; am = i; }
        }
        lastTag[b] = am;
    }
}

// ---------------------------------------------------------------------------
// Kernel 2: backtrack. One block per batch. TDM pulls bp[b] (256KB) into LDS
// so the 1024 dependent byte loads hit LDS instead of HBM.
// ---------------------------------------------------------------------------
__global__ __launch_bounds__(256) void viterbi_backward(
    const unsigned char* __restrict__ bp, const int* __restrict__ lengths,
    const int* __restrict__ lastTag, int* __restrict__ tags)
{
    extern __shared__ char smem[];
    unsigned char* sbp = (unsigned char*)smem;              // 256 KB
    int* stags = (int*)(smem + Tc * Nc);                    // 4 KB

    const int b = blockIdx.x;

    if (threadIdx.x == 0) {
        tdm_load_2d_to_lds((uint32)(size_t)sbp, bp + (size_t)b * Tc * Nc,
                           /*1B*/0, Nc, Tc, Nc, Nc, Tc);
    }
    __builtin_amdgcn_s_wait_tensorcnt(0);
    __syncthreads();

    if (threadIdx.x == 0) {
        const int len = lengths[b];
        int tag = lastTag[b];
        for (int t = Tc - 1; t >= 0; --t) {
            stags[t] = (t < len) ? tag : 0;                 // emit
            if (t >= 1 && t < len) tag = (int)sbp[t * Nc + tag];
        }
    }
    __syncthreads();

    // coalesced write-out: 256 threads x int4 = 4KB
    int4* dst = (int4*)(tags + (size_t)b * Tc);
    const int4* src = (const int4*)stags;
    dst[threadIdx.x] = src[threadIdx.x];
}

// ---------------------------------------------------------------------------
// Kernel 3: one-hot expansion, pure-bandwidth float4 stores over ALL of d_out
// (also overwrites the bp scratch region -> deterministic output).
// ---------------------------------------------------------------------------
__global__ __launch_bounds__(256) void onehot_kernel(
    const int* __restrict__ tags, float* __restrict__ out)
{
    size_t idx = (size_t)blockIdx.x * blockDim.x + threadIdx.x; // float4 index
    int row = (int)(idx >> 6);                                  // 64 float4 / row
    int qq  = (int)(idx & 63);
    int tag = tags[row];
    float4 v = make_float4(0.f, 0.f, 0.f, 0.f);
    int d = tag - (qq << 2);
    if (d >= 0 && d < 4) (&v.x)[d] = 1.0f;
    ((float4*)out)[idx] = v;
}

// ---------------------------------------------------------------------------
extern "C" void kernel_launch(void* const* d_in, const int* in_sizes, int n_in,
                              void* d_out, int out_size, void* d_ws, size_t ws_size,
                              hipStream_t stream)
{
    (void)in_sizes; (void)n_in; (void)out_size; (void)ws_size;
    const float* pot   = (const float*)d_in[0];   // [B,T,N] f32
    const float* trans = (const float*)d_in[1];   // [N,N]   f32
    const int*   lens  = (const int*)d_in[2];     // [B]     i32

    float* out = (float*)d_out;
    // Scratch: bp (u8, B*T*N = 33.5MB) lives in d_out (134MB) until kernel 3
    // overwrites everything. tags + lastTag live in d_ws.
    unsigned char* bp = (unsigned char*)d_out;
    int* tags    = (int*)d_ws;                                  // B*T ints (512KB)
    int* lastTag = (int*)((char*)d_ws + (size_t)Bc * Tc * 4);   // B ints

    const size_t lds_fwd = 262144 + 1024 + 4096 + 4096;         // 265 KB
    const size_t lds_bwd = (size_t)Tc * Nc + Tc * 4;            // 260 KB

    viterbi_forward<<<Bc, Nc, lds_fwd, stream>>>(pot, trans, lens, bp, lastTag);
    viterbi_backward<<<Bc, Nc, lds_bwd, stream>>>(bp, lens, lastTag, tags);

    const int total_f4 = Bc * Tc * (Nc / 4);                    // 8,388,608
    onehot_kernel<<<total_f4 / 256, 256, 0, stream>>>(tags, out);
}